// VariationalPatchKernel_65481071406604
// MI455X (gfx1250) — compile-verified
//
#include <hip/hip_runtime.h>
#include <math.h>

typedef float v2f __attribute__((ext_vector_type(2)));
typedef float v8f __attribute__((ext_vector_type(8)));

#define NLAYERS 3
#define NQ 4

// ---------------------------------------------------------------------------
// Setup kernel (1 wave): build the params-only 16x16 unitary U by simulating
// the layered circuit on the 16 basis states, then pack B = U^T (B[k][n] =
// U[n][k]) into the WMMA-B operand layout for V_WMMA_F32_16X16X4_F32:
// for K-slice q, lane l holds rows k0 = 4q + 2*(l>>4) and k0+1 at column
// n = l&15, two contiguous floats per lane. Re: ws[0..255], Im: ws[256..511].
// Uses fast hardware trig (|t/2| small); this wave runs serially before the
// main kernel, so keeping it short matters.
// ---------------------------------------------------------------------------
__global__ void vqc_build_unitary(const float* __restrict__ params,
                                  float* __restrict__ ws) {
  __shared__ float Ure[16][16]; // [column j][amplitude n] = U[n][j]
  __shared__ float Uim[16][16];
  const int tid = threadIdx.x; // 0..31
  if (tid < 16) {
    float cre[16], cim[16];
#pragma unroll
    for (int n = 0; n < 16; ++n) { cre[n] = (n == tid) ? 1.0f : 0.0f; cim[n] = 0.0f; }
    for (int l = 0; l < NLAYERS; ++l) {
#pragma unroll
      for (int w = 0; w < NQ; ++w) {
        const int mask = 8 >> w; // wire w <-> bit (3-w)
        { // RY(params[l][w][0])
          const float t = 0.5f * params[l * 8 + w * 2 + 0];
          const float c = __cosf(t), s = __sinf(t);
#pragma unroll
          for (int i = 0; i < 16; ++i) {
            if ((i & mask) == 0) {
              const int j = i | mask;
              const float ar = cre[i], ai = cim[i], br = cre[j], bi = cim[j];
              cre[i] = c * ar - s * br; cim[i] = c * ai - s * bi;
              cre[j] = s * ar + c * br; cim[j] = s * ai + c * bi;
            }
          }
        }
        { // RZ(params[l][w][1]) = diag(e^{-it/2}, e^{+it/2})
          const float t = 0.5f * params[l * 8 + w * 2 + 1];
          const float c = __cosf(t), s = __sinf(t);
#pragma unroll
          for (int i = 0; i < 16; ++i) {
            const float re = cre[i], im = cim[i];
            if ((i & mask) == 0) { cre[i] = re * c + im * s; cim[i] = im * c - re * s; }
            else                 { cre[i] = re * c - im * s; cim[i] = im * c + re * s; }
          }
        }
      }
#pragma unroll
      for (int w = 0; w < NQ - 1; ++w) { // CNOT(control=w, target=w+1)
        const int cm = 8 >> w, tm = 4 >> w;
#pragma unroll
        for (int i = 0; i < 16; ++i) {
          if ((i & cm) && !(i & tm)) {
            const int j = i | tm;
            const float r = cre[i], m2 = cim[i];
            cre[i] = cre[j]; cim[i] = cim[j];
            cre[j] = r;      cim[j] = m2;
          }
        }
      }
    }
#pragma unroll
    for (int n = 0; n < 16; ++n) { Ure[tid][n] = cre[n]; Uim[tid][n] = cim[n]; }
  }
  __syncthreads();
  const int n = tid & 15, h = tid >> 4;
#pragma unroll
  for (int q = 0; q < 4; ++q) {
    const int k0 = 4 * q + 2 * h;
    ws[(q * 32 + tid) * 2 + 0]       = Ure[k0][n];     // B[k0][n] = U[n][k0]
    ws[(q * 32 + tid) * 2 + 1]       = Ure[k0 + 1][n];
    ws[256 + (q * 32 + tid) * 2 + 0] = Uim[k0][n];
    ws[256 + (q * 32 + tid) * 2 + 1] = Uim[k0 + 1][n];
  }
}

// lane-XOR swap via DS_SWIZZLE_B32 (group-of-32 mode: and=0x1f, or=0, xor=d)
__device__ __forceinline__ float lane_xor(float x, int d) {
  const int imm = (d << 10) | 0x1f;
  int r;
  switch (d) { // immediate must be a constant
    case 1: r = __builtin_amdgcn_ds_swizzle(__float_as_int(x), (1 << 10) | 0x1f); break;
    case 2: r = __builtin_amdgcn_ds_swizzle(__float_as_int(x), (2 << 10) | 0x1f); break;
    case 4: r = __builtin_amdgcn_ds_swizzle(__float_as_int(x), (4 << 10) | 0x1f); break;
    default: r = __builtin_amdgcn_ds_swizzle(__float_as_int(x), (8 << 10) | 0x1f); break;
  }
  (void)imm;
  return __int_as_float(r);
}

// ---------------------------------------------------------------------------
// Main kernel: grid-stride loop, one 16-patch tile per wave per iteration.
// Both 16-lane halves compute the same patch's product-state vector v, so the
// WMMA-A 16x4 operand needs only a per-half register select (no permutes).
// 4 K-slices of v_wmma_f32_16x16x4_f32 x {Re,Im} give u = U v; square to
// probs; a 4-stage lane-XOR Walsh-Hadamard yields all four <Z_w> per row.
// B operands and lane constants are hoisted outside the tile loop.
// ---------------------------------------------------------------------------
__global__ __launch_bounds__(256)
void vqc_wmma_kernel(const float* __restrict__ patch,
                     const float* __restrict__ ws,
                     float* __restrict__ out, int ntiles) {
  const int lane  = threadIdx.x & 31;
  const int wave0 = (int)((blockIdx.x * blockDim.x + threadIdx.x) >> 5);
  const int nwaves = (int)(gridDim.x * (blockDim.x >> 5));
  const int h = lane >> 4;
  const int n = lane & 15;

  // B operands (uniform across all tiles): 4 K-slices x {Re, Im}
  const v2f* wsv = (const v2f*)ws;
  v2f bre[4], bim[4];
#pragma unroll
  for (int q = 0; q < 4; ++q) {
    bre[q] = wsv[q * 32 + lane];
    bim[q] = wsv[128 + q * 32 + lane];
  }

  // output slot for this lane: <Z_w> = WHT coeff at mask 8>>w
  int wsel = -1;
  if (n == 8) wsel = 0; else if (n == 4) wsel = 1;
  else if (n == 2) wsel = 2; else if (n == 1) wsel = 3;

  for (int tile = wave0; tile < ntiles; tile += nwaves) {
    const int m16 = tile * 16;

    // Per-patch product state v (duplicated across halves)
    const float4 t4 = ((const float4*)patch)[m16 + n];
    const float c0 = __cosf(0.5f * t4.x), s0 = __sinf(0.5f * t4.x);
    const float c1 = __cosf(0.5f * t4.y), s1 = __sinf(0.5f * t4.y);
    const float c2 = __cosf(0.5f * t4.z), s2 = __sinf(0.5f * t4.z);
    const float c3 = __cosf(0.5f * t4.w), s3 = __sinf(0.5f * t4.w);
    const float p01[4] = {c0 * c1, c0 * s1, s0 * c1, s0 * s1};
    const float p23[4] = {c2 * c3, c2 * s3, s2 * c3, s2 * s3};
    float v[16];
#pragma unroll
    for (int k = 0; k < 16; ++k) v[k] = p01[k >> 2] * p23[k & 3];

    v8f dre = {0.f, 0.f, 0.f, 0.f, 0.f, 0.f, 0.f, 0.f};
    v8f dim = {0.f, 0.f, 0.f, 0.f, 0.f, 0.f, 0.f, 0.f};
#pragma unroll
    for (int q = 0; q < 4; ++q) {
      v2f a;
      a.x = h ? v[4 * q + 2] : v[4 * q + 0]; // A layout: halves hold K, K+2
      a.y = h ? v[4 * q + 3] : v[4 * q + 1];
      dre = __builtin_amdgcn_wmma_f32_16x16x4_f32(false, a, false, bre[q],
                                                  (short)0, dre, false, false);
      dim = __builtin_amdgcn_wmma_f32_16x16x4_f32(false, a, false, bim[q],
                                                  (short)0, dim, false, false);
    }

    // probs in C/D layout: VGPR r, half h -> row m = r + 8h, column = n
    float p[8];
#pragma unroll
    for (int r = 0; r < 8; ++r) p[r] = dre[r] * dre[r] + dim[r] * dim[r];

    // Walsh-Hadamard butterfly over state-index lanes (d<16 stays within half)
#pragma unroll
    for (int d = 1; d <= 8; d <<= 1) {
#pragma unroll
      for (int r = 0; r < 8; ++r) {
        const float nb = lane_xor(p[r], d);
        p[r] = (lane & d) ? (nb - p[r]) : (p[r] + nb);
      }
    }

    if (wsel >= 0) {
#pragma unroll
      for (int r = 0; r < 8; ++r) {
        const int m = r + 8 * h;
        out[(m16 + m) * 4 + wsel] = p[r];
      }
    }
  }
}

extern "C" void kernel_launch(void* const* d_in, const int* in_sizes, int n_in,
                              void* d_out, int out_size, void* d_ws, size_t ws_size,
                              hipStream_t stream) {
  const float* patch  = (const float*)d_in[0];
  const float* params = (const float*)d_in[1];
  float* out = (float*)d_out;
  float* ws  = (float*)d_ws;

  const int B = in_sizes[0] / 4; // number of patches
  const int ntiles = B / 16;     // B = 2^20, divisible by 16

  vqc_build_unitary<<<1, 32, 0, stream>>>(params, ws);

  // ~4 tiles per wave: amortize B-operand loads and lane setup
  const int wavesPerBlock = 256 / 32;
  int blocks = (ntiles + wavesPerBlock * 4 - 1) / (wavesPerBlock * 4);
  if (blocks < 1) blocks = 1;
  vqc_wmma_kernel<<<blocks, 256, 0, stream>>>(patch, ws, out, ntiles);
}